// RoDAlignMax_4080218931274
// MI455X (gfx1250) — compile-verified
//
#include <hip/hip_runtime.h>
#include <stdint.h>

// Problem constants from the reference
#define B_    4
#define C_    256
#define H_    64
#define W_    64
#define AH_   10
#define AW_   10
#define GRID_ 100          // AH_*AW_
#define OQ_   81           // 9*9 pooled outputs per channel
#define CTILE_ 128         // channels per LDS buffer (51,200 B each, x2 buffers)

typedef float v4f __attribute__((ext_vector_type(4)));

// Use CDNA5 async global->LDS copies on gfx1250
#if defined(__gfx1250__)
#define ROD_ASYNC 1
#else
#define ROD_ASYNC 0
#endif

__device__ __forceinline__ void rod_wait_asynccnt_25() {
#if defined(__gfx1250__)
#if __has_builtin(__builtin_amdgcn_s_wait_asynccnt)
    __builtin_amdgcn_s_wait_asynccnt(25);
#else
    asm volatile("s_wait_asynccnt 25" ::: "memory");
#endif
#endif
}
__device__ __forceinline__ void rod_wait_asynccnt_0() {
#if defined(__gfx1250__)
#if __has_builtin(__builtin_amdgcn_s_wait_asynccnt)
    __builtin_amdgcn_s_wait_asynccnt(0);
#else
    asm volatile("s_wait_asynccnt 0" ::: "memory");
#endif
#endif
}

// ---------------------------------------------------------------------------
// Phase 1: samp[b][c][i][j] = features[b][c][7i][7j]
// (bilinear grid is exact: (H-1)/(ah-1) == 7.0, so interpolation == gather)
// ---------------------------------------------------------------------------
__global__ void rod_gather_kernel(const float* __restrict__ f,
                                  float* __restrict__ samp) {
    int idx = blockIdx.x * blockDim.x + threadIdx.x;
    if (idx >= B_ * C_ * GRID_) return;
    int ij = idx % GRID_;
    int bc = idx / GRID_;
    int i = ij / AW_;
    int j = ij - i * AW_;
    samp[idx] = f[bc * (H_ * W_) + (7 * i) * W_ + (7 * j)];
}

// ---------------------------------------------------------------------------
// Phase 2: one block per ROI. Double-buffered async staging of the batch slab
// into LDS, then masked 2x2/s1 max-pool with b128 nontemporal stores.
// out[r][c][qi][qj] = max over 2x2 of (inside ? 0 : samp[bidx][c][i][j])
// ---------------------------------------------------------------------------
__global__ __launch_bounds__(256)
void rod_pool_kernel(const float* __restrict__ samp,
                     const float* __restrict__ rois,
                     float* __restrict__ out) {
    __shared__ __align__(16) float lds[2][CTILE_ * GRID_];   // 102,400 B total

    const int r   = blockIdx.x;
    const int tid = threadIdx.x;

    const float bf = rois[r * 5 + 0];
    const float x1 = rois[r * 5 + 1] * 0.25f;
    const float y1 = rois[r * 5 + 2] * 0.25f;
    const float x2 = rois[r * 5 + 3] * 0.25f;
    const float y2 = rois[r * 5 + 4] * 0.25f;
    const int bidx = (int)bf;

    // 10-bit inside masks, exactly mirroring the reference float compares.
    unsigned in_h = 0u, in_w = 0u;
#pragma unroll
    for (int i = 0; i < AH_; ++i) {
        const float p = 7.0f * (float)i;          // exact grid coordinate
        in_h |= (p >= y1 && p <= y2) ? (1u << i) : 0u;
        in_w |= (p >= x1 && p <= x2) ? (1u << i) : 0u;
    }

    const float* sb = samp + (size_t)bidx * (C_ * GRID_);
    float*       ob = out + (size_t)r * (C_ * OQ_);

    // ---- issue async staging for BOTH halves up front (50 ops/wave < 63) ----
#if ROD_ASYNC
#pragma unroll 1
    for (int half = 0; half < 2; ++half) {
        const float* src = sb + half * (CTILE_ * GRID_);
        float*       dst = lds[half];
#pragma unroll 1
        for (int k = 0; k < (CTILE_ * GRID_) / (256 * 2); ++k) {   // 25 b64 ops
            const int v = k * 256 + tid;
            // Low 32 bits of a generic LDS pointer == LDS byte offset.
            unsigned lofs = (unsigned)(uintptr_t)(&dst[v * 2]);
            unsigned long long ga = (unsigned long long)(uintptr_t)(&src[v * 2]);
            asm volatile("global_load_async_to_lds_b64 %0, %1, off"
                         :: "v"(lofs), "v"(ga) : "memory");
        }
    }
    // Async loads complete in order: cnt<=25 => half 0 fully landed.
    rod_wait_asynccnt_25();
    __syncthreads();
#else
#pragma unroll 1
    for (int k = 0; k < (2 * CTILE_ * GRID_) / (256 * 2); ++k) {
        const int v = k * 256 + tid;
        ((float2*)&lds[0][0])[v] = ((const float2*)sb)[v];
    }
    __syncthreads();
#endif

    // ---- masked 2x2 stride-1 max-pool of one half, 4-wide NT stores ----
    auto pool_half = [&](const float* __restrict__ buf, float* __restrict__ obh) {
#pragma unroll 1
        for (int idx4 = tid; idx4 < (CTILE_ * OQ_) / 4; idx4 += 256) {
            v4f res;
#pragma unroll
            for (int u = 0; u < 4; ++u) {
                const int e  = idx4 * 4 + u;
                const int cl = e / OQ_;
                const int q  = e - cl * OQ_;
                const int i  = q / 9;
                const int j  = q - i * 9;
                const int base = cl * GRID_ + i * AW_ + j;

                const unsigned h0 = (in_h >> i) & 1u;
                const unsigned h1 = (in_h >> (i + 1)) & 1u;
                const unsigned w0 = (in_w >> j) & 1u;
                const unsigned w1 = (in_w >> (j + 1)) & 1u;

                const float v00 = (h0 & w0) ? 0.0f : buf[base];
                const float v01 = (h0 & w1) ? 0.0f : buf[base + 1];
                const float v10 = (h1 & w0) ? 0.0f : buf[base + AW_];
                const float v11 = (h1 & w1) ? 0.0f : buf[base + AW_ + 1];

                res[u] = fmaxf(fmaxf(v00, v01), fmaxf(v10, v11));
            }
            __builtin_nontemporal_store(res, (v4f*)(obh + idx4 * 4));
        }
    };

    pool_half(lds[0], ob);                       // half 0 (c in [0,128))

#if ROD_ASYNC
    rod_wait_asynccnt_0();                       // my half-1 writes landed
    __syncthreads();                             // everyone's landed
#endif

    pool_half(lds[1], ob + CTILE_ * OQ_);        // half 1 (c in [128,256))
}

// ---------------------------------------------------------------------------
extern "C" void kernel_launch(void* const* d_in, const int* in_sizes, int n_in,
                              void* d_out, int out_size, void* d_ws, size_t ws_size,
                              hipStream_t stream) {
    const float* features = (const float*)d_in[0];   // [4,256,64,64] f32
    const float* rois     = (const float*)d_in[1];   // [R,5] f32
    float*       out      = (float*)d_out;           // [R,256,9,9] f32
    float*       samp     = (float*)d_ws;            // [4,256,10,10] f32 = 409,600 B

    const int R = in_sizes[1] / 5;

    const int gthreads = B_ * C_ * GRID_;            // 102,400
    rod_gather_kernel<<<(gthreads + 255) / 256, 256, 0, stream>>>(features, samp);
    rod_pool_kernel<<<R, 256, 0, stream>>>(samp, rois, out);
}